// BinReg_17514876634182
// MI455X (gfx1250) — compile-verified
//
#include <hip/hip_runtime.h>

typedef __attribute__((ext_vector_type(2))) float v2f;
typedef __attribute__((ext_vector_type(4))) float v4f;
typedef __attribute__((ext_vector_type(8))) float v8f;

#define NBLOCKS 2048
#define NT      256
#define NBINS   16
#define NSLOT   49   // 16 cnt + 16 sum + 16 sumsq + 1 msq

__global__ __launch_bounds__(NT) void binreg_pass1(
    const float* __restrict__ w, const float* __restrict__ wq,
    const float* __restrict__ alpha, float* __restrict__ partial, int n)
{
    // [stat][bin][tid]: bin stride = 256 words == 0 mod 64 banks -> bank = tid -> conflict-free
    __shared__ float hist[3 * NBINS * NT];   // 48 KB
    __shared__ float msqbuf[NT];

    const int tid = threadIdx.x;

    // zero private slots
    #pragma unroll
    for (int s = 0; s < 3; ++s)
        #pragma unroll
        for (int b = 0; b < NBINS; ++b)
            hist[s * (NBINS * NT) + b * NT + tid] = 0.0f;
    __syncthreads();

    const float inv_a = 1.0f / alpha[0];
    float msq = 0.0f;

    float* h0 = &hist[tid];                    // counts
    float* h1 = &hist[NBINS * NT + tid];       // sum w
    float* h2 = &hist[2 * NBINS * NT + tid];   // sum w^2

    const int n4      = n >> 2;
    const int gstride = gridDim.x * blockDim.x;
    const int gid     = blockIdx.x * blockDim.x + tid;
    const v4f* w4  = (const v4f*)w;
    const v4f* wq4 = (const v4f*)wq;

    for (int i = gid; i < n4; i += gstride) {
        // streamed once, 512MB total >> 192MB L2: use non-temporal loads
        const v4f a = __builtin_nontemporal_load(&w4[i]);
        const v4f b = __builtin_nontemporal_load(&wq4[i]);
        #pragma unroll
        for (int c = 0; c < 4; ++c) {
            const float av = a[c];
            const float bv = b[c];
            const float d  = bv - av;
            msq += d * d;
            int bin = (int)__builtin_rintf(bv * inv_a) + 8;
            bin = bin < 0 ? 0 : (bin > 15 ? 15 : bin);
            atomicAdd(&h0[bin * NT], 1.0f);
            atomicAdd(&h1[bin * NT], av);
            atomicAdd(&h2[bin * NT], av * av);
        }
    }
    // scalar tail (n not multiple of 4)
    for (int i = (n4 << 2) + gid; i < n; i += gstride) {
        const float av = w[i];
        const float bv = wq[i];
        const float d  = bv - av;
        msq += d * d;
        int bin = (int)__builtin_rintf(bv * inv_a) + 8;
        bin = bin < 0 ? 0 : (bin > 15 ? 15 : bin);
        atomicAdd(&h0[bin * NT], 1.0f);
        atomicAdd(&h1[bin * NT], av);
        atomicAdd(&h2[bin * NT], av * av);
    }

    __syncthreads();
    msqbuf[tid] = msq;
    __syncthreads();
    // tree-reduce msq across the block
    for (int s = NT / 2; s > 0; s >>= 1) {
        if (tid < s) msqbuf[tid] += msqbuf[tid + s];
        __syncthreads();
    }

    // Reduce 16(bin) x 256(tid) per-stat arrays with WMMA: D = A(16x4 tile) * ones(4x16), chained C.
    const int wave = tid >> 5;
    const int lane = tid & 31;
    if (wave < 3) {
        const float* base = &hist[wave * (NBINS * NT)];
        const int row  = lane & 15;   // M = bin
        const int half = lane >> 4;   // lanes 16-31 supply K+2,K+3
        v2f ones; ones.x = 1.0f; ones.y = 1.0f;
        v8f acc = {};
        for (int k0 = 0; k0 < NT; k0 += 4) {
            v2f a;
            const int idx = row * NT + k0 + 2 * half;
            a.x = base[idx];
            a.y = base[idx + 1];
            acc = __builtin_amdgcn_wmma_f32_16x16x4_f32(
                false, a, false, ones, (short)0, acc, false, false);
        }
        // D layout: VGPR j, lanes 0-15 -> M=j, lanes 16-31 -> M=8+j (all N columns identical)
        float* outp = &partial[(size_t)blockIdx.x * NSLOT + wave * NBINS];
        if (lane == 0) {
            #pragma unroll
            for (int j = 0; j < 8; ++j) outp[j] = acc[j];
        } else if (lane == 16) {
            #pragma unroll
            for (int j = 0; j < 8; ++j) outp[8 + j] = acc[j];
        }
    }
    if (tid == 0) partial[(size_t)blockIdx.x * NSLOT + 48] = msqbuf[0];
}

__global__ __launch_bounds__(64) void binreg_pass2(
    const float* __restrict__ partial, float* __restrict__ out, int nblocks, int n)
{
    __shared__ double red[NSLOT];
    const int j = threadIdx.x;
    if (j < NSLOT) {
        double acc = 0.0;
        for (int b = 0; b < nblocks; ++b)      // fixed order -> deterministic
            acc += (double)partial[(size_t)b * NSLOT + j];
        red[j] = acc;
    }
    __syncthreads();
    if (j == 0) {
        double loss = red[48] / (double)n;     // mean((wq - w)^2)
        for (int bin = 0; bin < NBINS; ++bin) {
            const double cnt = red[bin];
            const double s   = red[NBINS + bin];
            const double ss  = red[2 * NBINS + bin];
            if (cnt > 1.0)                      // torch.var gating, denom max()s implied
                loss += (ss - s * s / cnt) / (cnt - 1.0);
        }
        out[0] = (float)(0.1 * loss);           // LMBDA
    }
}

extern "C" void kernel_launch(void* const* d_in, const int* in_sizes, int n_in,
                              void* d_out, int out_size, void* d_ws, size_t ws_size,
                              hipStream_t stream) {
    const float* w     = (const float*)d_in[0];
    const float* wq    = (const float*)d_in[1];
    const float* alpha = (const float*)d_in[3];   // d_in[2] = nbit (compile-time 4 here)
    const int n = in_sizes[0];

    int nb = NBLOCKS;
    const size_t need_per_block = NSLOT * sizeof(float);
    if ((size_t)nb * need_per_block > ws_size) {
        nb = (int)(ws_size / need_per_block);
        if (nb < 1) nb = 1;
    }
    float* partial = (float*)d_ws;

    binreg_pass1<<<nb, NT, 0, stream>>>(w, wq, alpha, partial, n);
    binreg_pass2<<<1, 64, 0, stream>>>(partial, (float*)d_out, nb, n);
}